// LayerIn2LayerOut_43404939493622
// MI455X (gfx1250) — compile-verified
//
#include <hip/hip_runtime.h>

#define D_FEAT        128
#define SCATTER_BLOCK 256
#define EDGE_CHUNK    512   // edges staged into LDS per block

// ---------------------------------------------------------------------------
// CDNA5 helpers: async global->LDS copy (ASYNCcnt-tracked) with safe fallback
// ---------------------------------------------------------------------------
typedef int cdna5_v4i __attribute__((vector_size(16)));

__device__ __forceinline__ void cdna5_async_ld_b128(const int* gsrc, int* lds_dst) {
#if defined(__gfx1250__) && __has_builtin(__builtin_amdgcn_global_load_async_to_lds_b128)
  __builtin_amdgcn_global_load_async_to_lds_b128(
      (__attribute__((address_space(1))) cdna5_v4i*)(cdna5_v4i*)(void*)gsrc,
      (__attribute__((address_space(3))) cdna5_v4i*)(cdna5_v4i*)(void*)lds_dst,
      /*imm offset*/ 0, /*cpol*/ 0);
#else
  // Fallback: normal load + LDS store (DScnt path); still correct.
  *(int4*)lds_dst = *(const int4*)gsrc;
#endif
}

__device__ __forceinline__ void cdna5_wait_async0() {
#if defined(__gfx1250__) && __has_builtin(__builtin_amdgcn_s_wait_asynccnt)
  __builtin_amdgcn_s_wait_asynccnt(0);
#elif defined(__gfx1250__)
  asm volatile("s_wait_asynccnt 0" ::: "memory");
#endif
}

// Relaxed agent-scope fp32 add: selects the native no-return
// global_atomic_add_f32 on gfx1250 (accumulator is device memory, so agent
// scope is sufficient and avoids any cmpswap emulation path).
__device__ __forceinline__ void atomic_add_f32_agent(float* p, float v) {
  (void)__hip_atomic_fetch_add(p, v, __ATOMIC_RELAXED, __HIP_MEMORY_SCOPE_AGENT);
}

// ---------------------------------------------------------------------------
// Scatter-add (defined FIRST so the disasm snippet shows it).
// One wave32 per edge row: lane l handles features [4l, 4l+3], so 32 lanes
// cover all 128 features with a single float4 gather. Edge indices are
// bulk-staged into LDS with the CDNA5 async copy engine (ASYNCcnt path).
// ds_in (51.2 MB) and the accumulator (51.2 MB) both fit in the 192 MB L2,
// so the 320 MB of random gathers and 80M fp32 atomics run at L2 rate while
// HBM only sees compulsory traffic (~160 MB -> ~7 us floor at 23.3 TB/s).
// ---------------------------------------------------------------------------
__global__ void __launch_bounds__(SCATTER_BLOCK) gnn_scatter_kernel(
    const float* __restrict__ ds_in, const int* __restrict__ edges,
    float* __restrict__ out, float* __restrict__ deg, int n_edges) {
  __shared__ __align__(16) int s_recv[EDGE_CHUNK];
  __shared__ __align__(16) int s_src[EDGE_CHUNK];

  const int chunk_start = blockIdx.x * EDGE_CHUNK;
  int cnt = n_edges - chunk_start;
  if (cnt > EDGE_CHUNK) cnt = EDGE_CHUNK;

  // Stage this block's slice of both index rows (recv row 0, src row 1) into
  // LDS. Full 16B quads use the async ASYNCcnt path; ragged tail uses guarded
  // scalar copies.
  const int t4 = threadIdx.x * 4;
  if (t4 + 4 <= cnt) {
    cdna5_async_ld_b128(edges + chunk_start + t4,           s_recv + t4);
    cdna5_async_ld_b128(edges + n_edges + chunk_start + t4, s_src  + t4);
  } else {
    for (int k = 0; k < 4; ++k) {
      int e = t4 + k;
      if (e < cnt) {
        s_recv[e] = edges[chunk_start + e];
        s_src[e]  = edges[n_edges + chunk_start + e];
      }
    }
  }
  cdna5_wait_async0();
  __syncthreads();

  const int lane   = threadIdx.x & 31;
  const int wave   = threadIdx.x >> 5;
  const int nwaves = SCATTER_BLOCK / 32;

  for (int e = wave; e < cnt; e += nwaves) {
    // Prefetch the next row this wave will gather (random access -> hide
    // L2/HBM latency with global_prefetch_b8).
    int en = e + nwaves;
    if (en < cnt) {
      __builtin_prefetch(ds_in + (size_t)s_src[en] * D_FEAT + lane * 4, 0, 1);
    }
    const int src = s_src[e];
    const int rcv = s_recv[e];
    const float4 v =
        reinterpret_cast<const float4*>(ds_in + (size_t)src * D_FEAT)[lane];
    float* row = out + (size_t)rcv * D_FEAT + lane * 4;
    atomic_add_f32_agent(row + 0, v.x);
    atomic_add_f32_agent(row + 1, v.y);
    atomic_add_f32_agent(row + 2, v.z);
    atomic_add_f32_agent(row + 3, v.w);
    if (lane == 0) atomic_add_f32_agent(deg + rcv, 1.0f);
  }
}

// ---------------------------------------------------------------------------
// Kernel 1: out = ds_out (float4 streaming), deg = 0
// ---------------------------------------------------------------------------
__global__ void __launch_bounds__(256) gnn_init_kernel(
    const float* __restrict__ ds_out, float* __restrict__ out,
    float* __restrict__ deg, int n_vec4, int n_nodes) {
  int i = blockIdx.x * blockDim.x + threadIdx.x;
  if (i < n_vec4) {
    reinterpret_cast<float4*>(out)[i] =
        reinterpret_cast<const float4*>(ds_out)[i];
  }
  if (i < n_nodes) deg[i] = 0.0f;
}

// ---------------------------------------------------------------------------
// Kernel 3: out[v,:] *= 1 / (deg[v] + 1)
// ---------------------------------------------------------------------------
__global__ void __launch_bounds__(256) gnn_norm_kernel(
    float* __restrict__ out, const float* __restrict__ deg, int n_vec4) {
  int i = blockIdx.x * blockDim.x + threadIdx.x;
  if (i < n_vec4) {
    int v = i >> 5;  // 32 float4 groups per 128-feature node
    float s = 1.0f / (deg[v] + 1.0f);
    float4 x = reinterpret_cast<float4*>(out)[i];
    x.x *= s; x.y *= s; x.z *= s; x.w *= s;
    reinterpret_cast<float4*>(out)[i] = x;
  }
}

// ---------------------------------------------------------------------------
extern "C" void kernel_launch(void* const* d_in, const int* in_sizes, int n_in,
                              void* d_out, int out_size, void* d_ws, size_t ws_size,
                              hipStream_t stream) {
  const float* ds_in  = (const float*)d_in[0];
  const float* ds_out = (const float*)d_in[1];
  const int*   edges  = (const int*)d_in[2];   // [2, E] row-major int32
  float* out = (float*)d_out;
  float* deg = (float*)d_ws;                   // n_nodes floats of scratch

  const int n_nodes = in_sizes[0] / D_FEAT;
  const int n_edges = in_sizes[2] / 2;
  const int n_vec4  = n_nodes * (D_FEAT / 4);

  {
    int grid = (n_vec4 + 255) / 256;
    hipLaunchKernelGGL(gnn_init_kernel, dim3(grid), dim3(256), 0, stream,
                       ds_out, out, deg, n_vec4, n_nodes);
  }
  {
    int grid = (n_edges + EDGE_CHUNK - 1) / EDGE_CHUNK;
    hipLaunchKernelGGL(gnn_scatter_kernel, dim3(grid), dim3(SCATTER_BLOCK), 0,
                       stream, ds_in, edges, out, deg, n_edges);
  }
  {
    int grid = (n_vec4 + 255) / 256;
    hipLaunchKernelGGL(gnn_norm_kernel, dim3(grid), dim3(256), 0, stream,
                       out, deg, n_vec4);
  }
}